// PosEncoding_fix_2714419331615
// MI455X (gfx1250) — compile-verified
//
#include <hip/hip_runtime.h>

// Sinusoidal positional encoding, MI455X (gfx1250).
// out[row, i] = (i even) ? sin(pos[row] * w_{i/2}) : cos(pos[row] * w_{i/2})
//   w_j = 10000^(-j/256),  D = 512;  pos==0 -> all-zero row.
//
// Memory-bound: 512MB fp32 output >> 192MB L2 -> stream with non-temporal
// b128 stores; trans math on the hardware v_sin/v_cos/v_exp units.

typedef float v4f __attribute__((ext_vector_type(4)));  // clang vector: OK for nontemporal builtins

namespace {
constexpr unsigned kD             = 512;
constexpr unsigned kThreadsPerRow = 64;   // each thread emits 8 floats (4 pairs)
constexpr unsigned kBlock         = 256;  // 8 wave32s per block
}

__global__ __launch_bounds__(256) void pe_sincos_kernel(
    const float* __restrict__ positions,
    float* __restrict__       out,
    unsigned                  n_rows)
{
    const unsigned g    = blockIdx.x * kBlock + threadIdx.x;
    const unsigned row  = g >> 6;        // wave-uniform (64 threads per row)
    const unsigned slot = g & 63u;       // which 8-float chunk of the row
    if (row >= n_rows) return;

    const float p     = positions[row];
    const float zmask = (p == 0.0f) ? 0.0f : 1.0f;

    // Pair index for the first of this thread's 4 sin/cos pairs.
    const float j0 = (float)(slot * 4u);

    // w_j = 10000^(-j/256) = exp2(-j * log2(10000)/256)
    const float kC = 0.051905126482615036f;  // log2(10000)/256
    const float w0 = __builtin_amdgcn_exp2f(-j0 * kC);  // v_exp_f32
    // Geometric ratio between consecutive pairs: r = 10000^(-1/256)
    const float r1 = 0.96466162f;
    const float r2 = 0.93057206f;  // r^2
    const float r3 = 0.89768716f;  // r^3
    const float w1 = w0 * r1;
    const float w2 = w0 * r2;
    const float w3 = w0 * r3;

    // v_sin_f32 / v_cos_f32 take input in revolutions (arg * 2*pi).
    const float kInv2Pi = 0.15915494309189535f;
    const float pr = p * kInv2Pi;
    const float a0 = pr * w0;
    const float a1 = pr * w1;
    const float a2 = pr * w2;
    const float a3 = pr * w3;

    v4f v0, v1;
    v0.x = zmask * __builtin_amdgcn_sinf(a0);
    v0.y = zmask * __builtin_amdgcn_cosf(a0);
    v0.z = zmask * __builtin_amdgcn_sinf(a1);
    v0.w = zmask * __builtin_amdgcn_cosf(a1);
    v1.x = zmask * __builtin_amdgcn_sinf(a2);
    v1.y = zmask * __builtin_amdgcn_cosf(a2);
    v1.z = zmask * __builtin_amdgcn_sinf(a3);
    v1.w = zmask * __builtin_amdgcn_cosf(a3);

    // 32-byte-aligned destination; two b128 non-temporal streaming stores.
    float* dst = out + (size_t)row * kD + (size_t)slot * 8u;
    __builtin_nontemporal_store(v0, (v4f*)dst);
    __builtin_nontemporal_store(v1, (v4f*)(dst + 4));
}

extern "C" void kernel_launch(void* const* d_in, const int* in_sizes, int n_in,
                              void* d_out, int out_size, void* d_ws, size_t ws_size,
                              hipStream_t stream) {
    (void)n_in; (void)out_size; (void)d_ws; (void)ws_size;
    const float* positions = (const float*)d_in[0];
    float*       out       = (float*)d_out;
    const unsigned n_rows  = (unsigned)in_sizes[0];

    const unsigned total_threads = n_rows * kThreadsPerRow;
    const unsigned blocks        = (total_threads + kBlock - 1) / kBlock;
    pe_sincos_kernel<<<blocks, kBlock, 0, stream>>>(positions, out, n_rows);
}